// TGAT_68607807586902
// MI455X (gfx1250) — compile-verified
//
#include <hip/hip_runtime.h>
#include <hip/hip_bf16.h>

#define NNODES 100000
#define NEDGES 800000

typedef __attribute__((ext_vector_type(2))) float v2f;
typedef __attribute__((ext_vector_type(8))) float v8f;

// ---------- monotone float <-> uint encoding for atomic max ----------
__device__ __forceinline__ unsigned f2ord(float f) {
    unsigned u = __float_as_uint(f);
    return (u & 0x80000000u) ? ~u : (u | 0x80000000u);
}
__device__ __forceinline__ float ord2f(unsigned e) {
    unsigned u = (e & 0x80000000u) ? (e ^ 0x80000000u) : ~e;
    return __uint_as_float(u);
}

// ---------- zero / init accumulators ----------
__global__ __launch_bounds__(256) void k_zero(float* __restrict__ htime,
                                              float* __restrict__ deg,
                                              float* __restrict__ s,
                                              unsigned* __restrict__ menc, int n) {
    int i = blockIdx.x * blockDim.x + threadIdx.x;
    if (i < n * 128) htime[i] = 0.0f;
    if (i < n) deg[i] = 0.0f;
    if (i < n * 4) { s[i] = 0.0f; menc[i] = 0u; }  // ord-encoded -max
}

// ---------- fused time-encoding + u_add_e + segment_sum (wave per edge) ----------
__global__ __launch_bounds__(256) void k_time_scatter(
    const float* __restrict__ xin, const float* __restrict__ time,
    const int* __restrict__ src, const int* __restrict__ dst,
    const float* __restrict__ wt, const float* __restrict__ bt,
    float* __restrict__ htime, float* __restrict__ deg, int E) {
    int tid = blockIdx.x * blockDim.x + threadIdx.x;
    int e = tid >> 5, lane = tid & 31;
    if (e >= E) return;
    int si = src[e], di = dst[e];
    float t = time[e];
    int c = lane * 4;
    float4 xh = *(const float4*)(xin + (size_t)si * 128 + c);
    float4 w  = *(const float4*)(wt + c);
    float4 b  = *(const float4*)(bt + c);
    float v0 = xh.x + __cosf(t * w.x + b.x);
    float v1 = xh.y + __cosf(t * w.y + b.y);
    float v2 = xh.z + __cosf(t * w.z + b.z);
    float v3 = xh.w + __cosf(t * w.w + b.w);
    float* o = htime + (size_t)di * 128 + c;
    atomicAdd(o + 0, v0); atomicAdd(o + 1, v1);
    atomicAdd(o + 2, v2); atomicAdd(o + 3, v3);
    if (lane == 0) atomicAdd(deg + di, 1.0f);
}

// ---------- mean-normalize h_time ----------
__global__ __launch_bounds__(256) void k_norm(float* __restrict__ htime,
                                              const float* __restrict__ deg, int n) {
    int i = blockIdx.x * blockDim.x + threadIdx.x;
    if (i >= n * 128) return;
    htime[i] /= fmaxf(deg[i >> 7], 1.0f);
}

// ---------- WMMA fp32 GEMM: C[M x (NT*16)] = [A0|A1][M x K] @ W^T + bias ----------
// A panels are row-major, row stride 128. W is [NT*16, K] row-major.
// W is staged through LDS in 64-wide K slices using async global->LDS copies.
template <int NT>
__global__ __launch_bounds__(256) void k_gemm_wmma(
    const float* __restrict__ A0, const float* __restrict__ A1,
    const float* __restrict__ W, const float* __restrict__ bias,
    float* __restrict__ C, int M, int K, int K0, int ldc, int relu) {
    __shared__ float sW[NT * 16 * 64];  // [cols][64] K-slice, <=32KB

    int lane = threadIdx.x & 31;
    int waveInBlk = threadIdx.x >> 5;
    int row0 = (blockIdx.x * 8 + waveInBlk) * 16;
    bool active = row0 < M;             // wave-uniform
    int lnlo = lane & 15;
    int hi   = lane >> 4;
    int rowA = row0 + lnlo;

    v8f acc[NT];
#pragma unroll
    for (int t = 0; t < NT; ++t) acc[t] = (v8f)0.0f;

    const int totalV4 = NT * 16 * 64 / 4;  // float4 chunks in the slice (mult of 256)

    for (int k0 = 0; k0 < K; k0 += 64) {
        // ---- async cooperative stage: W[:, k0:k0+64) -> LDS ----
        for (int idx = threadIdx.x; idx < totalV4; idx += 256) {
            int c  = (idx * 4) >> 6;   // column (output feature)
            int kk = (idx * 4) & 63;   // k within slice
            unsigned ldsoff = (unsigned)(unsigned long long)(sW + idx * 4);
            unsigned long long ga =
                (unsigned long long)(W + (size_t)c * K + k0 + kk);
            asm volatile("global_load_async_to_lds_b128 %0, %1, off"
                         :: "v"(ldsoff), "v"(ga) : "memory");
        }
        asm volatile("s_wait_asynccnt 0x0" ::: "memory");
        __syncthreads();

        if (active) {
            for (int kk = 0; kk < 64; kk += 4) {
                int k = k0 + kk;
                const float* Ap; int ka;
                if (k < K0) { Ap = A0; ka = k; } else { Ap = A1; ka = k - K0; }
                // A 16x4 frag: lane row M=lnlo, v0=A[M][k+2*hi], v1=A[M][k+2*hi+1]
                v2f a = *(const v2f*)(Ap + (size_t)rowA * 128 + ka + 2 * hi);
                // batch all B frags from LDS, then issue the WMMA chain
                v2f b[NT];
#pragma unroll
                for (int t = 0; t < NT; ++t)
                    b[t] = *(const v2f*)(sW + (t * 16 + lnlo) * 64 + kk + 2 * hi);
#pragma unroll
                for (int t = 0; t < NT; ++t)
                    acc[t] = __builtin_amdgcn_wmma_f32_16x16x4_f32(
                        false, a, false, b[t], (short)0, acc[t], false, false);
            }
        }
        __syncthreads();
    }

    if (!active) return;
#pragma unroll
    for (int t = 0; t < NT; ++t) {
        int col = t * 16 + lnlo;
        float bv = bias ? bias[col] : 0.0f;
#pragma unroll
        for (int r = 0; r < 8; ++r) {
            int row = row0 + r + 8 * hi;  // C layout: VGPR r -> M=r / M=r+8
            float v = acc[t][r] + bv;
            if (relu) v = fmaxf(v, 0.0f);
            C[(size_t)row * ldc + col] = v;
        }
    }
}

// ---------- per-(node,head) attention logits el/er ----------
__global__ __launch_bounds__(256) void k_attn_el_er(
    const float* __restrict__ feat, const float* __restrict__ al,
    const float* __restrict__ ar, float* __restrict__ el,
    float* __restrict__ er, int n) {
    int idx = blockIdx.x * blockDim.x + threadIdx.x;
    if (idx >= n * 4) return;
    int node = idx >> 2, h = idx & 3;
    const float* fr = feat + (size_t)node * 128 + h * 32;
    const float* pl = al + h * 32;
    const float* pr = ar + h * 32;
    float accl = 0.0f, accr = 0.0f;
#pragma unroll
    for (int j = 0; j < 32; j += 4) {
        float4 f = *(const float4*)(fr + j);
        float4 wl = *(const float4*)(pl + j);
        float4 wr = *(const float4*)(pr + j);
        accl += f.x * wl.x + f.y * wl.y + f.z * wl.z + f.w * wl.w;
        accr += f.x * wr.x + f.y * wr.y + f.z * wr.z + f.w * wr.w;
    }
    el[idx] = accl;
    er[idx] = accr;
}

// ---------- edge scores + segment max (thread per edge) ----------
__global__ __launch_bounds__(256) void k_edge_score(
    const float* __restrict__ el, const float* __restrict__ er,
    const int* __restrict__ src, const int* __restrict__ dst,
    float* __restrict__ esc, unsigned* __restrict__ menc, int E) {
    int e = blockIdx.x * blockDim.x + threadIdx.x;
    if (e >= E) return;
    int si = src[e], di = dst[e];
    float4 a = *(const float4*)(el + (size_t)si * 4);
    float4 b = *(const float4*)(er + (size_t)di * 4);
    float e0 = a.x + b.x; e0 = (e0 >= 0.0f) ? e0 : 0.2f * e0;
    float e1 = a.y + b.y; e1 = (e1 >= 0.0f) ? e1 : 0.2f * e1;
    float e2 = a.z + b.z; e2 = (e2 >= 0.0f) ? e2 : 0.2f * e2;
    float e3 = a.w + b.w; e3 = (e3 >= 0.0f) ? e3 : 0.2f * e3;
    float4 o; o.x = e0; o.y = e1; o.z = e2; o.w = e3;
    *(float4*)(esc + (size_t)e * 4) = o;
    unsigned* m = menc + (size_t)di * 4;
    atomicMax(m + 0, f2ord(e0)); atomicMax(m + 1, f2ord(e1));
    atomicMax(m + 2, f2ord(e2)); atomicMax(m + 3, f2ord(e3));
}

// ---------- exp(e - m) + segment sum ----------
__global__ __launch_bounds__(256) void k_exp_sum(
    float* __restrict__ esc, const unsigned* __restrict__ menc,
    const int* __restrict__ dst, float* __restrict__ s, int E) {
    int e = blockIdx.x * blockDim.x + threadIdx.x;
    if (e >= E) return;
    int di = dst[e];
    float4 ev = *(const float4*)(esc + (size_t)e * 4);
    uint4 mm = *(const uint4*)(menc + (size_t)di * 4);
    float e0 = __expf(ev.x - ord2f(mm.x));
    float e1 = __expf(ev.y - ord2f(mm.y));
    float e2 = __expf(ev.z - ord2f(mm.z));
    float e3 = __expf(ev.w - ord2f(mm.w));
    float4 o; o.x = e0; o.y = e1; o.z = e2; o.w = e3;
    *(float4*)(esc + (size_t)e * 4) = o;
    float* sp = s + (size_t)di * 4;
    atomicAdd(sp + 0, e0); atomicAdd(sp + 1, e1);
    atomicAdd(sp + 2, e2); atomicAdd(sp + 3, e3);
}

// ---------- alpha-weighted message scatter (wave per edge) ----------
__global__ __launch_bounds__(256) void k_alpha_scatter(
    const float* __restrict__ esc, const float* __restrict__ s,
    const float* __restrict__ feat, const int* __restrict__ src,
    const int* __restrict__ dst, float* __restrict__ outacc, int E) {
    int tid = blockIdx.x * blockDim.x + threadIdx.x;
    int e = tid >> 5, lane = tid & 31;
    if (e >= E) return;
    int si = src[e], di = dst[e];
    int h = lane >> 3;  // 8 lanes (32 cols) per head
    float alpha = esc[(size_t)e * 4 + h] / s[(size_t)di * 4 + h];
    int c = lane * 4;
    float4 f = *(const float4*)(feat + (size_t)si * 128 + c);
    float* o = outacc + (size_t)di * 128 + c;
    atomicAdd(o + 0, alpha * f.x); atomicAdd(o + 1, alpha * f.y);
    atomicAdd(o + 2, alpha * f.z); atomicAdd(o + 3, alpha * f.w);
}

// ---------- ELU ----------
__global__ __launch_bounds__(256) void k_elu(const float* __restrict__ in,
                                             float* __restrict__ out, int total) {
    int i = blockIdx.x * blockDim.x + threadIdx.x;
    if (i >= total) return;
    float v = in[i];
    out[i] = (v > 0.0f) ? v : expm1f(v);
}

// ---------- final tiny classifier layer (OUT/2 -> NC) ----------
__global__ __launch_bounds__(256) void k_logits(
    const float* __restrict__ c1, const float* __restrict__ Wc2,
    const float* __restrict__ bc2, float* __restrict__ out, int n) {
    int node = blockIdx.x * blockDim.x + threadIdx.x;
    if (node >= n) return;
    float a0 = bc2[0], a1 = bc2[1];
    const float* r = c1 + (size_t)node * 64;
#pragma unroll
    for (int j = 0; j < 64; j += 4) {
        float4 v = *(const float4*)(r + j);
        float4 w0 = *(const float4*)(Wc2 + j);
        float4 w1 = *(const float4*)(Wc2 + 64 + j);
        a0 += v.x * w0.x + v.y * w0.y + v.z * w0.z + v.w * w0.w;
        a1 += v.x * w1.x + v.y * w1.y + v.z * w1.z + v.w * w1.w;
    }
    out[(size_t)node * 2 + 0] = a0;
    out[(size_t)node * 2 + 1] = a1;
}

// =====================================================================
extern "C" void kernel_launch(void* const* d_in, const int* in_sizes, int n_in,
                              void* d_out, int out_size, void* d_ws, size_t ws_size,
                              hipStream_t stream) {
    const int N = NNODES, E = NEDGES;
    const float* h    = (const float*)d_in[0];
    const float* time = (const float*)d_in[1];
    const int*   src  = (const int*)d_in[2];
    const int*   dst  = (const int*)d_in[3];
    const float* Wp   = (const float*)d_in[4];
    const float* bp   = (const float*)d_in[5];
    const float* wt[2]   = {(const float*)d_in[6],  (const float*)d_in[13]};
    const float* bt[2]   = {(const float*)d_in[7],  (const float*)d_in[14]};
    const float* Wfc[2]  = {(const float*)d_in[8],  (const float*)d_in[15]};
    const float* al[2]   = {(const float*)d_in[9],  (const float*)d_in[16]};
    const float* ar[2]   = {(const float*)d_in[10], (const float*)d_in[17]};
    const float* Wres[2] = {(const float*)d_in[11], (const float*)d_in[18]};
    const float* bg[2]   = {(const float*)d_in[12], (const float*)d_in[19]};
    const float* Wc1 = (const float*)d_in[20];
    const float* bc1 = (const float*)d_in[21];
    const float* Wc2 = (const float*)d_in[22];
    const float* bc2 = (const float*)d_in[23];
    float* out = (float*)d_out;

    // workspace carve-up
    float* P0 = (float*)d_ws;          // [N,128] layer input / output (ping)
    float* P1 = P0 + (size_t)N * 128;  // [N,128] h_time (reused as c1 [N,64])
    float* P2 = P1 + (size_t)N * 128;  // [N,128] feat
    float* P3 = P2 + (size_t)N * 128;  // [N,128] out accumulator (resid init)
    float* el = P3 + (size_t)N * 128;  // [N,4]
    float* er = el + (size_t)N * 4;    // [N,4]
    float* deg = er + (size_t)N * 4;   // [N]
    unsigned* menc = (unsigned*)(deg + N);      // [N,4]
    float* s  = (float*)(menc + (size_t)N * 4); // [N,4]
    float* esc = s + (size_t)N * 4;    // [E,4]

    const int TPB = 256;
    int gNode128 = (N * 128 + TPB - 1) / TPB;       // 50000
    int gNodeH   = (N * 4 + TPB - 1) / TPB;
    int gEdge    = (E + TPB - 1) / TPB;
    int gEdgeWave = (E * 32 + TPB - 1) / TPB;       // wave per edge
    int waves    = N / 16;                          // 6250 row tiles
    int gGemm    = (waves + 7) / 8;                 // 8 waves per block

    // x0 = h @ Wp^T + bp
    k_gemm_wmma<8><<<gGemm, TPB, 0, stream>>>(h, nullptr, Wp, bp, P0,
                                              N, 128, 128, 128, 0);

    for (int li = 0; li < 2; ++li) {
        k_zero<<<gNode128, TPB, 0, stream>>>(P1, deg, s, menc, N);
        k_time_scatter<<<gEdgeWave, TPB, 0, stream>>>(P0, time, src, dst,
                                                      wt[li], bt[li], P1, deg, E);
        k_norm<<<gNode128, TPB, 0, stream>>>(P1, deg, N);
        // feat = [x | h_time] @ Wfc^T
        k_gemm_wmma<8><<<gGemm, TPB, 0, stream>>>(P0, P1, Wfc[li], nullptr, P2,
                                                  N, 256, 128, 128, 0);
        // out_acc = [x | h_time] @ Wres^T + bg   (residual initializes accumulator)
        k_gemm_wmma<8><<<gGemm, TPB, 0, stream>>>(P0, P1, Wres[li], bg[li], P3,
                                                  N, 256, 128, 128, 0);
        k_attn_el_er<<<gNodeH, TPB, 0, stream>>>(P2, al[li], ar[li], el, er, N);
        k_edge_score<<<gEdge, TPB, 0, stream>>>(el, er, src, dst, esc, menc, E);
        k_exp_sum<<<gEdge, TPB, 0, stream>>>(esc, menc, dst, s, E);
        k_alpha_scatter<<<gEdgeWave, TPB, 0, stream>>>(esc, s, P2, src, dst, P3, E);
        k_elu<<<gNode128, TPB, 0, stream>>>(P3, P0, N * 128);  // next layer input
    }

    // classifier: c1 = relu(h2 @ Wc1^T + bc1), logits = c1 @ Wc2^T + bc2
    k_gemm_wmma<4><<<gGemm, TPB, 0, stream>>>(P0, nullptr, Wc1, bc1, P1,
                                              N, 128, 128, 64, 1);
    int gN = (N + TPB - 1) / TPB;
    k_logits<<<gN, TPB, 0, stream>>>(P1, Wc2, bc2, out, N);
}